// InteractiveAttention_33380485825331
// MI455X (gfx1250) — compile-verified
//
#include <hip/hip_runtime.h>
#include <math.h>

// Problem dims (fixed by the reference)
constexpr int Bn  = 4;
constexpr int LAn = 512;
constexpr int LOn = 512;
constexpr int Hn  = 256;

// Tiling for the interaction kernel
constexpr int TO = 32;   // opinion rows per LDS tile (32*256*4 = 32 KB)
constexpr int TA = 4;    // aspect rows sharing each po tile per block
constexpr int NT = LOn / TO;

typedef float v2f __attribute__((ext_vector_type(2)));
typedef float v8f __attribute__((ext_vector_type(8)));
typedef int   v4i __attribute__((ext_vector_type(4)));

// ---- hardware tanh (gfx1250 TRANS op) --------------------------------------
__device__ __forceinline__ float fast_tanhf(float x) {
#if __has_builtin(__builtin_amdgcn_tanhf)
  return __builtin_amdgcn_tanhf(x);
#elif __has_builtin(__builtin_amdgcn_tanh_f32)
  return __builtin_amdgcn_tanh_f32(x);
#else
  float r;
  asm volatile("v_tanh_f32 %0, %1" : "=v"(r) : "v"(x));
  return r;
#endif
}

// ---- async global->LDS copy (ASYNCcnt path, bypasses VGPRs) ----------------
#if __has_builtin(__builtin_amdgcn_global_load_async_to_lds_b128) && \
    __has_builtin(__builtin_amdgcn_s_wait_asynccnt)
#define HAVE_ASYNC_LDS 1
#else
#define HAVE_ASYNC_LDS 0
#endif

// Issue one TO*Hn-float tile copy (32 KB): 8 x b128 per thread, coalesced.
__device__ __forceinline__ void issue_tile_load(const float* __restrict__ src,
                                                float* dstLds, int t) {
#if HAVE_ASYNC_LDS
#pragma unroll
  for (int i = 0; i < (TO * Hn / 4) / 256; ++i) {
    const int idx = (t + i * 256) * 4;   // float index, 16B granules
    __builtin_amdgcn_global_load_async_to_lds_b128(
        (__attribute__((address_space(1))) v4i*)(src + idx),
        (__attribute__((address_space(3))) v4i*)(dstLds + idx),
        0, 0);
  }
#else
  const float4* s4 = (const float4*)src;
  float4*       d4 = (float4*)dstLds;
#pragma unroll
  for (int i = t; i < TO * Hn / 4; i += 256) d4[i] = s4[i];
#endif
}

__device__ __forceinline__ void wait_tile_loads() {
#if HAVE_ASYNC_LDS
  __builtin_amdgcn_s_wait_asynccnt(0);
#endif
}

// ---------------------------------------------------------------------------
// Kernel 1: Out[M, 256] = X[M, 256] * W1[:, wOff:wOff+256]^T
// One wave owns a 16x16 f32 tile; K-loop via V_WMMA_F32_16X16X4_F32.
//   A frag (16x4): lanes 0-15 -> M=lane, K=0/1 in v0/v1; lanes 16-31 -> K=2/3.
//   B frag (4x16): mirrored, N on lanes: B[k][n] = W1[n][wOff+h+k] (= W^T).
//   C/D: vreg v -> M=v (lanes 0-15) / M=v+8 (lanes 16-31), N=lane&15.
// ---------------------------------------------------------------------------
__global__ void __launch_bounds__(32)
proj_wmma_f32(const float* __restrict__ X, const float* __restrict__ W1,
              float* __restrict__ Out, int wOff)
{
  const int lane = threadIdx.x & 31;
  const int n16  = lane & 15;
  const int hi   = lane >> 4;
  const int colBase = blockIdx.x << 4;
  const int rowBase = blockIdx.y << 4;

  const float* xrow = X  + (size_t)(rowBase + n16) * Hn      + 2 * hi;
  const float* wrow = W1 + (size_t)(colBase + n16) * (2*Hn) + wOff + 2 * hi;

  v8f c = {0.f, 0.f, 0.f, 0.f, 0.f, 0.f, 0.f, 0.f};

#pragma unroll 8
  for (int h = 0; h < Hn; h += 4) {
    v2f a = *(const v2f*)(xrow + h);
    v2f b = *(const v2f*)(wrow + h);
    c = __builtin_amdgcn_wmma_f32_16x16x4_f32(false, a, false, b,
                                              (short)0, c, false, false);
  }

  float* orow = Out + (size_t)rowBase * Hn + colBase + n16;
#pragma unroll
  for (int v = 0; v < 8; ++v)
    orow[(size_t)(v + 8 * hi) * Hn] = c[v];
}

// ---------------------------------------------------------------------------
// Kernel 2: scores[a,o] = sum_k tanh(pa[a,k] + po[o,k] + b1[k]) * w2[k] + b2
// then softmax over o. hidden tensor lives only in registers.
// po streamed through double-buffered LDS tiles via async global->LDS loads;
// k is lane-strided (lane + 32*i): 32 lanes -> 32 distinct LDS banks.
// ---------------------------------------------------------------------------
__global__ void __launch_bounds__(256)
attn_tanh_softmax(const float* __restrict__ pa, const float* __restrict__ po,
                  const float* __restrict__ b1, const float* __restrict__ W2,
                  const float* __restrict__ b2, float* __restrict__ out)
{
  __shared__ float sPo[2][TO * Hn];    // 64 KB ping-pong
  __shared__ float sPa[TA * Hn];       //  4 KB
  __shared__ float sB1[Hn];
  __shared__ float sW2[Hn];
  __shared__ float sScores[TA * LOn];  //  8 KB
  __shared__ float sRed[8];

  const int t    = threadIdx.x;
  const int wave = t >> 5;
  const int lane = t & 31;

  const int blockRow = blockIdx.x * TA;     // first flat (b*LA + a) row
  const int b        = blockRow / LAn;

  {
    const float* paBase = pa + (size_t)blockRow * Hn;
#pragma unroll
    for (int i = t; i < TA * Hn; i += 256) sPa[i] = paBase[i];
    sB1[t] = b1[t];
    sW2[t] = W2[t];
  }
  const float  bias2 = b2[0];
  const float* poB   = po + (size_t)b * LOn * Hn;

  issue_tile_load(poB, sPo[0], t);          // prime the pipeline

  for (int tile = 0; tile < NT; ++tile) {
    const int cur = tile & 1;
    wait_tile_loads();                      // own async loads arrived
    __syncthreads();                        // everyone's arrived + prev compute done
    if (tile + 1 < NT)
      issue_tile_load(poB + (size_t)(tile + 1) * TO * Hn, sPo[cur ^ 1], t);

    const float* bufPo = sPo[cur];
#pragma unroll
    for (int j = 0; j < TO / 8; ++j) {      // 8 waves x (TO/8) o's
      const int    oL   = wave * (TO / 8) + j;
      const float* prow = bufPo + oL * Hn;
#pragma unroll
      for (int a = 0; a < TA; ++a) {
        const float* parow = sPa + a * Hn;
        float s = 0.f;
#pragma unroll
        for (int i = 0; i < Hn / 32; ++i) {
          const int k = lane + 32 * i;
          s += fast_tanhf(parow[k] + prow[k] + sB1[k]) * sW2[k];
        }
#pragma unroll
        for (int off = 16; off; off >>= 1) s += __shfl_xor(s, off, 32);
        if (lane == 0) sScores[a * LOn + tile * TO + oL] = s + bias2;
      }
    }
  }
  __syncthreads();

  // per-row softmax over LO=512 (2 elements per thread)
  for (int a = 0; a < TA; ++a) {
    const float* row = sScores + a * LOn;
    const float x0 = row[t], x1 = row[t + 256];

    float m = fmaxf(x0, x1);
#pragma unroll
    for (int off = 16; off; off >>= 1) m = fmaxf(m, __shfl_xor(m, off, 32));
    if (lane == 0) sRed[wave] = m;
    __syncthreads();
    m = sRed[0];
#pragma unroll
    for (int w = 1; w < 8; ++w) m = fmaxf(m, sRed[w]);
    __syncthreads();                        // sRed about to be reused

    const float e0 = __expf(x0 - m), e1 = __expf(x1 - m);
    float s = e0 + e1;
#pragma unroll
    for (int off = 16; off; off >>= 1) s += __shfl_xor(s, off, 32);
    if (lane == 0) sRed[wave] = s;
    __syncthreads();
    float sum = 0.f;
#pragma unroll
    for (int w = 0; w < 8; ++w) sum += sRed[w];
    const float inv = 1.0f / sum;

    float* orow = out + ((size_t)blockRow + a) * LOn;
    orow[t]       = e0 * inv;
    orow[t + 256] = e1 * inv;
    __syncthreads();                        // before next row reuses sRed
  }
}

// ---------------------------------------------------------------------------
extern "C" void kernel_launch(void* const* d_in, const int* in_sizes, int n_in,
                              void* d_out, int out_size, void* d_ws, size_t ws_size,
                              hipStream_t stream)
{
  const float* aspect  = (const float*)d_in[0];  // (4,512,256)
  const float* opinion = (const float*)d_in[1];  // (4,512,256)
  const float* W1      = (const float*)d_in[2];  // (256,512)
  const float* b1      = (const float*)d_in[3];  // (256,)
  const float* W2      = (const float*)d_in[4];  // (1,256)
  const float* b2      = (const float*)d_in[5];  // (1,)
  float* out = (float*)d_out;                    // (4,512,512)

  float* pa = (float*)d_ws;                      // 2 MB
  float* po = pa + (size_t)Bn * LAn * Hn;        // 2 MB

  dim3 gA(Hn / 16, (Bn * LAn) / 16);             // (16, 128) wavetiles
  proj_wmma_f32<<<gA, 32, 0, stream>>>(aspect,  W1, pa, 0);
  dim3 gO(Hn / 16, (Bn * LOn) / 16);
  proj_wmma_f32<<<gO, 32, 0, stream>>>(opinion, W1, po, Hn);

  attn_tanh_softmax<<<(Bn * LAn) / TA, 256, 0, stream>>>(pa, po, b1, W2, b2, out);
}